// GraphCut_60327110640306
// MI455X (gfx1250) — compile-verified
//
#include <hip/hip_runtime.h>
#include <hip/hip_bf16.h>

typedef float v2f __attribute__((ext_vector_type(2)));
typedef float v8f __attribute__((ext_vector_type(8)));

#define N_ROWS 8192
#define KDIM   128
#define NTILE  64            // 8192 / 128 tiles per dim
#define LDS_STRIDE 36        // 32 K-cols + 4 pad: rows 16B-aligned, banks spread

__global__ void row_norms_kernel(const float* __restrict__ X,
                                 float* __restrict__ norms) {
    int i = blockIdx.x * blockDim.x + threadIdx.x;
    if (i < N_ROWS) {
        const float4* row = (const float4*)(X + (size_t)i * KDIM);
        float s = 0.f;
#pragma unroll
        for (int q = 0; q < KDIM / 4; ++q) {
            float4 v = row[q];
            s += v.x * v.x + v.y * v.y + v.z * v.z + v.w * v.w;
        }
        norms[i] = s;
    }
}

__global__ __launch_bounds__(256)
void graphcut_tile_kernel(const float* __restrict__ X,
                          const int* __restrict__ Y,
                          const float* __restrict__ norms,
                          float* __restrict__ out) {
    __shared__ float sA[128 * LDS_STRIDE];
    __shared__ float sB[128 * LDS_STRIDE];
    __shared__ float red[256];

    const int t  = threadIdx.x;
    const int w  = t >> 5;       // wave 0..7 -> M-strip w*16..w*16+15
    const int l  = t & 31;       // lane
    const int m  = l & 15;       // A row-in-subtile / B col-in-subtile
    const int kh = l >> 4;       // K-half select (fp32 16x16x4 layout)

    // linear block id -> upper-triangular tile (bi <= bj)
    int bi = 0, rem = (int)blockIdx.x;
    while (rem >= NTILE - bi) { rem -= NTILE - bi; ++bi; }
    const int bj = bi + rem;
    const int i0 = bi * 128, j0 = bj * 128;
    const float scale = (bi == bj) ? 1.0f : 2.0f;   // off-diag tiles counted twice

    v8f acc[8] = {};

    for (int kc = 0; kc < KDIM / 32; ++kc) {
        __syncthreads();
        // stage 128x32 chunk of each of the two row-blocks into LDS
#pragma unroll
        for (int q = 0; q < 4; ++q) {
            int idx = t + 256 * q;               // 0..1023 float4 slots per tile
            int row = idx >> 3;
            int c4  = (idx & 7) * 4;
            *(float4*)&sA[row * LDS_STRIDE + c4] =
                *(const float4*)&X[(size_t)(i0 + row) * KDIM + kc * 32 + c4];
            *(float4*)&sB[row * LDS_STRIDE + c4] =
                *(const float4*)&X[(size_t)(j0 + row) * KDIM + kc * 32 + c4];
        }
        __syncthreads();
#pragma unroll
        for (int k4 = 0; k4 < 8; ++k4) {
            int col = k4 * 4 + kh * 2;           // this lane's 2 K-elements
            v2f a = *(const v2f*)&sA[(w * 16 + m) * LDS_STRIDE + col];
#pragma unroll
            for (int s = 0; s < 8; ++s) {
                v2f b = *(const v2f*)&sB[(s * 16 + m) * LDS_STRIDE + col];
                acc[s] = __builtin_amdgcn_wmma_f32_16x16x4_f32(
                    false, a, false, b, (short)0, acc[s], false, false);
            }
        }
    }

    // epilogue: D/C layout -> vgpr r: lanes0-15 M=r, lanes16-31 M=r+8; N = l&15
    const int half8 = kh * 8;
    float niv[8]; int yiv[8];
#pragma unroll
    for (int r = 0; r < 8; ++r) {
        int i = i0 + w * 16 + r + half8;
        niv[r] = norms[i];
        yiv[r] = Y[i];
    }

    float local = 0.f;
#pragma unroll
    for (int s = 0; s < 8; ++s) {
        int j = j0 + s * 16 + m;
        float nj = norms[j];
        int  yj = Y[j];
#pragma unroll
        for (int r = 0; r < 8; ++r) {
            float g  = acc[s][r];
            float sq = niv[r] + nj - 2.0f * g;
            float d  = (sq > 0.f) ? sqrtf(sq) : 0.f;
            local += (yiv[r] == yj ? 1.0f : -0.5f) * d;
        }
    }
    local *= scale;

    red[t] = local;
    __syncthreads();
    for (int off = 128; off > 0; off >>= 1) {
        if (t < off) red[t] += red[t + off];
        __syncthreads();
    }
    if (t == 0) atomicAdd(out, red[0]);
}

extern "C" void kernel_launch(void* const* d_in, const int* in_sizes, int n_in,
                              void* d_out, int out_size, void* d_ws, size_t ws_size,
                              hipStream_t stream) {
    const float* X = (const float*)d_in[0];
    const int*   Y = (const int*)d_in[1];
    float* out   = (float*)d_out;
    float* norms = (float*)d_ws;          // 8192 floats = 32 KB scratch

    hipMemsetAsync(d_out, 0, (size_t)out_size * sizeof(float), stream);

    row_norms_kernel<<<N_ROWS / 256, 256, 0, stream>>>(X, norms);

    const int nblocks = NTILE * (NTILE + 1) / 2;   // 2080 upper-tri tiles
    graphcut_tile_kernel<<<nblocks, 256, 0, stream>>>(X, Y, norms, out);
}